// spikeSqure0_3_75780402970642
// MI455X (gfx1250) — compile-verified
//
#include <hip/hip_runtime.h>

// 16-step elementwise spiking recurrence, fused + division-eliminated.
//
//   reference per step:  v -= z*h[t]; z = heaviside((v - T[t]) / (|v|+1)); out += z*d[t]
//   |v|+1 > 0 for all finite v, so sign((v-T)/(|v|+1)) == sign(v-T)  =>  z = (v > T[t])
//
// Per step per element: 1 fma (v update), 1 v_cmp, 1 v_cndmask (z=1.0/0.0),
// 1 fma (out update) -> 64 VALU ops/element (the op-count floor for direct eval).
// VALU-bound by ~4x over the 23.3 TB/s memory floor, so the kernel exposes
// 8 independent per-thread chains (v8f) to maximize wave32 VOPD dual-issue
// pairing, and uses non-temporal b128 loads/stores + global_prefetch_b8 for
// the touch-once 512 MB stream.

typedef float v8f __attribute__((ext_vector_type(8)));

__global__ __launch_bounds__(256) void spike_recur16_kernel(
    const float* __restrict__ x, float* __restrict__ out,
    unsigned long long n8, unsigned long long n)
{
    // Fixed constants from the reference (fp32 round-trip exact).
    constexpr float H[16] = {
        -0.45029134f, 0.13660802f, 0.79116625f, 0.35154283f,
         0.2300263f,  0.25131857f, 0.2859055f,  0.35303813f,
         0.09955073f, 0.16905762f, 0.43071112f, 0.46997476f,
         0.2730424f,  0.13731983f, 0.30262393f, -0.5251225f };
    constexpr float D[16] = {
         0.29946914f, 0.01354178f, 1.3328267f,  0.58273536f,
         1.2930123f,  1.5332928f,  1.523752f,   0.44793314f,
         0.5867976f,  0.9837378f,  0.8790002f,  0.45363167f,
         0.22681575f, 0.2594645f,  0.07598496f, 0.12070131f };
    constexpr float T[16] = {
         1.1091447f, -0.999358f,   0.9960277f,  0.35394624f,
         1.0220968f,  0.9840779f,  0.78471875f, 0.52230215f,
         0.48891294f, 0.41686916f, 0.27483794f, 0.04275339f,
        -0.14358011f, -0.23020536f, -0.57749504f, 0.10610177f };

    const v8f* __restrict__ xv = reinterpret_cast<const v8f*>(x);
    v8f* __restrict__ ov       = reinterpret_cast<v8f*>(out);

    const unsigned long long stride =
        (unsigned long long)gridDim.x * blockDim.x;
    unsigned long long i =
        (unsigned long long)blockIdx.x * blockDim.x + threadIdx.x;

    for (; i < n8; i += stride) {
        // Prefetch next grid-stride chunk (global_prefetch_b8).
        if (i + stride < n8)
            __builtin_prefetch(reinterpret_cast<const void*>(xv + i + stride), 0, 1);

        // Streaming 2x128-bit non-temporal load: data is touched exactly once.
        v8f v = __builtin_nontemporal_load(xv + i);
        v8f z = {0.0f, 0.0f, 0.0f, 0.0f, 0.0f, 0.0f, 0.0f, 0.0f};
        v8f o = {0.0f, 0.0f, 0.0f, 0.0f, 0.0f, 0.0f, 0.0f, 0.0f};

#pragma unroll
        for (int t = 0; t < 16; ++t) {
            v = v - z * H[t];                        // v_fma (z from previous step)
#pragma unroll
            for (int c = 0; c < 8; ++c)
                z[c] = (v[c] > T[t]) ? 1.0f : 0.0f;  // v_cmp_gt + v_cndmask
            o = o + z * D[t];                        // v_fma accumulate output
        }

        __builtin_nontemporal_store(o, ov + i);      // streaming 2x128-bit store
    }

    // Scalar tail (n not divisible by 8) — no-op for the reference shape.
    const unsigned long long tail_start = n8 * 8ull;
    for (unsigned long long j = tail_start +
             (unsigned long long)blockIdx.x * blockDim.x + threadIdx.x;
         j < n; j += stride) {
        float v = x[j], z = 0.0f, o = 0.0f;
#pragma unroll
        for (int t = 0; t < 16; ++t) {
            v = v - z * H[t];
            z = (v > T[t]) ? 1.0f : 0.0f;
            o = o + z * D[t];
        }
        out[j] = o;
    }
}

extern "C" void kernel_launch(void* const* d_in, const int* in_sizes, int n_in,
                              void* d_out, int out_size, void* d_ws, size_t ws_size,
                              hipStream_t stream) {
    const float* x = (const float*)d_in[0];
    float* out     = (float*)d_out;

    const unsigned long long n  = (unsigned long long)in_sizes[0]; // 64*1024*1024
    const unsigned long long n8 = n / 8ull;

    const int threads = 256;          // 8 wave32 waves per block
    const unsigned long long ipt = 2; // v8f chunks per thread target
    unsigned long long blocks =
        (n8 + (unsigned long long)threads * ipt - 1) /
        ((unsigned long long)threads * ipt);
    if (blocks < 1) blocks = 1;
    if (blocks > 0x7FFFFFFFull) blocks = 0x7FFFFFFFull;

    spike_recur16_kernel<<<dim3((unsigned)blocks), dim3(threads), 0, stream>>>(
        x, out, n8, n);
}